// SkipGAT_64647847739808
// MI455X (gfx1250) — compile-verified
//
#include <hip/hip_runtime.h>
#include <math.h>

// GATv2 hetero conv for MI455X (gfx1250, wave32).
// Projections via V_WMMA_F32_16X16X4_F32 (exact fp32 on the matrix pipe);
// edge passes are wave-per-edge coalesced gather/scatter with atomic
// segment-max / segment-sum for the per-destination softmax.

typedef float v2f __attribute__((ext_vector_type(2)));
typedef float v8f __attribute__((ext_vector_type(8)));

// ---------------------------------------------------------------- utilities
__global__ void __launch_bounds__(256) fill_kernel(float* __restrict__ p, float v, size_t n) {
  size_t i = (size_t)blockIdx.x * blockDim.x + threadIdx.x;
  if (i < n) p[i] = v;
}

__device__ __forceinline__ void atomicMaxF(float* addr, float val) {
  // IEEE-ordered atomic max: int max for non-negative, unsigned min for negative.
  if (val >= 0.0f) atomicMax((int*)addr, __float_as_int(val));
  else             atomicMin((unsigned int*)addr, (unsigned int)__float_as_int(val));
}

// ------------------------------------------------- projection GEMM (WMMA f32)
// Y[N,128] = X[N,64] @ W[64,128].  One block = 16 rows; 8 waves cover the 8
// column tiles of 16.  K=64 -> 16 chained V_WMMA_F32_16X16X4_F32 per wave.
__global__ void __launch_bounds__(256) gemm_proj(const float* __restrict__ X,
                                                 const float* __restrict__ W,
                                                 float* __restrict__ Y, int N) {
  __shared__ float Xs[16 * 66];           // padded stride 66: no bank conflicts
  const int m0 = blockIdx.x * 16;

  // Cooperative, coalesced load of the 16x64 X tile into LDS.
  for (int i = threadIdx.x; i < 16 * 64; i += 256) {
    int r = i >> 6, c = i & 63;
    long gr = (long)(m0 + r);
    if (gr >= N) gr = (long)N - 1;        // clamp reads; stores are guarded
    Xs[r * 66 + c] = X[gr * 64 + c];
  }
  __syncthreads();

  const int lane  = threadIdx.x & 31;
  const int wave  = threadIdx.x >> 5;     // 0..7 -> column tile
  const int n0    = wave * 16;
  const int l15   = lane & 15;            // A row within tile / B,C column
  const int khalf = lane >> 4;            // K sub-pair select (0 -> K0/K1, 1 -> K2/K3)

  v8f acc = {};
#pragma unroll
  for (int k0 = 0; k0 < 64; k0 += 4) {
    const int kb = k0 + khalf * 2;
    // A fragment: lane holds X[m0+l15][kb], X[m0+l15][kb+1]  (ISA 16x4 f32 layout)
    v2f a = *(const v2f*)&Xs[l15 * 66 + kb];
    // B fragment: lane holds W[kb][n0+l15], W[kb+1][n0+l15]
    v2f b;
    b.x = W[(size_t)kb * 128 + n0 + l15];
    b.y = W[(size_t)(kb + 1) * 128 + n0 + l15];
    acc = __builtin_amdgcn_wmma_f32_16x16x4_f32(false, a, false, b,
                                                (short)0, acc, false, false);
  }

  // C/D layout: VGPR v -> row m0 + v + 8*(lane>=16), col n0 + l15.
#pragma unroll
  for (int v = 0; v < 8; ++v) {
    int m = m0 + v + ((lane >= 16) ? 8 : 0);
    if (m < N) Y[(size_t)m * 128 + n0 + l15] = acc[v];
  }
}

// ------------------------------------------------------- edge pass 1: logits
// One wave per edge; lane l covers channel l of each of the 4 heads
// (indices h*32+l -> four fully coalesced 128B chunks per feature row).
__global__ void __launch_bounds__(256) edge_alpha(const float* __restrict__ xl,
                                                  const float* __restrict__ xr,
                                                  const int* __restrict__ src,
                                                  const int* __restrict__ dst,
                                                  const float* __restrict__ att,
                                                  float* __restrict__ alpha,
                                                  float* __restrict__ amax, int E) {
  const int lane = threadIdx.x & 31;
  const int eid  = blockIdx.x * 8 + (threadIdx.x >> 5);
  if (eid >= E) return;
  const int s = src[eid], d = dst[eid];
  const float* pl = xl + (size_t)s * 128;
  const float* pr = xr + (size_t)d * 128;

  float a[4];
#pragma unroll
  for (int h = 0; h < 4; ++h) {
    float m = pl[h * 32 + lane] + pr[h * 32 + lane];
    float e = (m >= 0.0f) ? m : 0.2f * m;          // LeakyReLU(0.2)
    a[h] = e * att[h * 32 + lane];
  }
  // Full-wave butterfly reduction: every lane ends with all four head sums.
#pragma unroll
  for (int off = 16; off > 0; off >>= 1) {
#pragma unroll
    for (int h = 0; h < 4; ++h) a[h] += __shfl_xor(a[h], off, 32);
  }
  if (lane < 4) {
    alpha[(size_t)eid * 4 + lane] = a[lane];
    atomicMaxF(&amax[(size_t)d * 4 + lane], a[lane]);
  }
}

// ---------------------------------------------------- edge pass 2: softmax denom
__global__ void __launch_bounds__(256) edge_denom(const float* __restrict__ alpha,
                                                  const int* __restrict__ dst,
                                                  const float* __restrict__ amax,
                                                  float* __restrict__ denom, int E) {
  size_t idx = (size_t)blockIdx.x * blockDim.x + threadIdx.x;
  if (idx >= (size_t)E * 4) return;
  size_t e = idx >> 2;
  int h = (int)(idx & 3);
  int d = dst[e];
  float ex = expf(alpha[idx] - amax[(size_t)d * 4 + h]);
  atomicAdd(&denom[(size_t)d * 4 + h], ex);
}

// ------------------------------------------------- edge pass 3: weighted scatter
__global__ void __launch_bounds__(256) edge_scatter(const float* __restrict__ xl,
                                                    const float* __restrict__ alpha,
                                                    const int* __restrict__ src,
                                                    const int* __restrict__ dst,
                                                    const float* __restrict__ amax,
                                                    const float* __restrict__ denom,
                                                    float* __restrict__ out, int E) {
  const int lane = threadIdx.x & 31;
  const int eid  = blockIdx.x * 8 + (threadIdx.x >> 5);
  if (eid >= E) return;
  const int s = src[eid], d = dst[eid];

  float a[4];
#pragma unroll
  for (int h = 0; h < 4; ++h) {
    float ex = expf(alpha[(size_t)eid * 4 + h] - amax[(size_t)d * 4 + h]);
    a[h] = ex / (denom[(size_t)d * 4 + h] + 1e-16f);
  }
  const float* pl = xl + (size_t)s * 128;
  float* po = out + (size_t)d * 128;
#pragma unroll
  for (int h = 0; h < 4; ++h)
    atomicAdd(&po[h * 32 + lane], pl[h * 32 + lane] * a[h]);
}

// ---------------------------------------------------------------- bias epilogue
__global__ void __launch_bounds__(256) bias_add(float* __restrict__ out,
                                                const float* __restrict__ b_tt,
                                                const float* __restrict__ b_bt,
                                                const float* __restrict__ b_tb,
                                                int NTX, int NBD) {
  size_t i = (size_t)blockIdx.x * blockDim.x + threadIdx.x;
  size_t total = (size_t)(NTX + NBD) * 128;
  if (i >= total) return;
  int c = (int)(i & 127);
  size_t r = i >> 7;
  out[i] += (r < (size_t)NTX) ? (b_tt[c] + b_bt[c]) : b_tb[c];
}

// ------------------------------------------------------------------ launcher
extern "C" void kernel_launch(void* const* d_in, const int* in_sizes, int n_in,
                              void* d_out, int out_size, void* d_ws, size_t ws_size,
                              hipStream_t stream) {
  const float* x_tx  = (const float*)d_in[0];
  const float* x_bd  = (const float*)d_in[1];
  const int* src_tt  = (const int*)d_in[2];
  const int* dst_tt  = (const int*)d_in[3];
  const int* src_tb  = (const int*)d_in[4];
  const int* dst_tb  = (const int*)d_in[5];
  const int* src_bt  = (const int*)d_in[6];
  const int* dst_bt  = (const int*)d_in[7];
  const float* Wl_tt = (const float*)d_in[8];
  const float* Wr_tt = (const float*)d_in[9];
  const float* att_tt= (const float*)d_in[10];
  const float* b_tt  = (const float*)d_in[11];
  const float* Wl_tb = (const float*)d_in[12];
  const float* Wr_tb = (const float*)d_in[13];
  const float* att_tb= (const float*)d_in[14];
  const float* b_tb  = (const float*)d_in[15];
  const float* Wl_bt = (const float*)d_in[16];
  const float* Wr_bt = (const float*)d_in[17];
  const float* att_bt= (const float*)d_in[18];
  const float* b_bt  = (const float*)d_in[19];

  const int NTX = in_sizes[0] / 64;
  const int NBD = in_sizes[1] / 64;
  const int ETT = in_sizes[2];
  const int ETB = in_sizes[4];
  const int EBT = in_sizes[6];

  float* out = (float*)d_out;

  // ---- workspace carve-up (floats)
  float* ws = (float*)d_ws;
  size_t o = 0;
  float* xl_tt = ws + o; o += (size_t)NTX * 128;   // x_tx @ Wl_tt
  float* xr_tt = ws + o; o += (size_t)NTX * 128;   // x_tx @ Wr_tt
  float* xl_tb = ws + o; o += (size_t)NTX * 128;   // x_tx @ Wl_tb (src side)
  float* xr_tb = ws + o; o += (size_t)NBD * 128;   // x_bd @ Wr_tb (dst side)
  float* xl_bt = ws + o; o += (size_t)NBD * 128;   // x_bd @ Wl_bt (src side)
  float* xr_bt = ws + o; o += (size_t)NTX * 128;   // x_tx @ Wr_bt (dst side)
  float* al_tt = ws + o; o += (size_t)ETT * 4;
  float* al_tb = ws + o; o += (size_t)ETB * 4;
  float* al_bt = ws + o; o += (size_t)EBT * 4;
  float* amax_tt = ws + o; o += (size_t)NTX * 4;
  float* amax_tb = ws + o; o += (size_t)NBD * 4;
  float* amax_bt = ws + o; o += (size_t)NTX * 4;
  float* den_tt  = ws + o; o += (size_t)NTX * 4;
  float* den_tb  = ws + o; o += (size_t)NBD * 4;
  float* den_bt  = ws + o; o += (size_t)NTX * 4;

  const unsigned B = 256;
  auto gblk = [](size_t n, unsigned b) { return (unsigned)((n + b - 1) / b); };

  // ---- init (every call: harness poisons buffers)
  size_t nOut  = (size_t)(NTX + NBD) * 128;
  size_t nAmax = ((size_t)NTX + NBD + NTX) * 4;   // amax_tt..amax_bt contiguous
  hipLaunchKernelGGL(fill_kernel, dim3(gblk(nOut, B)),  dim3(B), 0, stream, out, 0.0f, nOut);
  hipLaunchKernelGGL(fill_kernel, dim3(gblk(nAmax, B)), dim3(B), 0, stream, amax_tt, -INFINITY, nAmax);
  hipLaunchKernelGGL(fill_kernel, dim3(gblk(nAmax, B)), dim3(B), 0, stream, den_tt, 0.0f, nAmax);

  // ---- projections (WMMA f32, exact)
  unsigned gTX = (unsigned)((NTX + 15) / 16), gBD = (unsigned)((NBD + 15) / 16);
  hipLaunchKernelGGL(gemm_proj, dim3(gTX), dim3(B), 0, stream, x_tx, Wl_tt, xl_tt, NTX);
  hipLaunchKernelGGL(gemm_proj, dim3(gTX), dim3(B), 0, stream, x_tx, Wr_tt, xr_tt, NTX);
  hipLaunchKernelGGL(gemm_proj, dim3(gTX), dim3(B), 0, stream, x_tx, Wl_tb, xl_tb, NTX);
  hipLaunchKernelGGL(gemm_proj, dim3(gBD), dim3(B), 0, stream, x_bd, Wr_tb, xr_tb, NBD);
  hipLaunchKernelGGL(gemm_proj, dim3(gBD), dim3(B), 0, stream, x_bd, Wl_bt, xl_bt, NBD);
  hipLaunchKernelGGL(gemm_proj, dim3(gTX), dim3(B), 0, stream, x_tx, Wr_bt, xr_bt, NTX);

  // ---- edge passes, per relation (wave-per-edge: 8 edges / 256-thread block)
  // tt: tx -> tx
  hipLaunchKernelGGL(edge_alpha,   dim3(gblk(ETT, 8)),              dim3(B), 0, stream,
                     xl_tt, xr_tt, src_tt, dst_tt, att_tt, al_tt, amax_tt, ETT);
  hipLaunchKernelGGL(edge_denom,   dim3(gblk((size_t)ETT * 4, B)),  dim3(B), 0, stream,
                     al_tt, dst_tt, amax_tt, den_tt, ETT);
  hipLaunchKernelGGL(edge_scatter, dim3(gblk(ETT, 8)),              dim3(B), 0, stream,
                     xl_tt, al_tt, src_tt, dst_tt, amax_tt, den_tt, out, ETT);
  // bt: bd -> tx (sums into same tx rows)
  hipLaunchKernelGGL(edge_alpha,   dim3(gblk(EBT, 8)),              dim3(B), 0, stream,
                     xl_bt, xr_bt, src_bt, dst_bt, att_bt, al_bt, amax_bt, EBT);
  hipLaunchKernelGGL(edge_denom,   dim3(gblk((size_t)EBT * 4, B)),  dim3(B), 0, stream,
                     al_bt, dst_bt, amax_bt, den_bt, EBT);
  hipLaunchKernelGGL(edge_scatter, dim3(gblk(EBT, 8)),              dim3(B), 0, stream,
                     xl_bt, al_bt, src_bt, dst_bt, amax_bt, den_bt, out, EBT);
  // tb: tx -> bd (writes bd rows after the tx block)
  float* out_bd = out + (size_t)NTX * 128;
  hipLaunchKernelGGL(edge_alpha,   dim3(gblk(ETB, 8)),              dim3(B), 0, stream,
                     xl_tb, xr_tb, src_tb, dst_tb, att_tb, al_tb, amax_tb, ETB);
  hipLaunchKernelGGL(edge_denom,   dim3(gblk((size_t)ETB * 4, B)),  dim3(B), 0, stream,
                     al_tb, dst_tb, amax_tb, den_tb, ETB);
  hipLaunchKernelGGL(edge_scatter, dim3(gblk(ETB, 8)),              dim3(B), 0, stream,
                     xl_tb, al_tb, src_tb, dst_tb, amax_tb, den_tb, out_bd, ETB);

  // ---- biases
  hipLaunchKernelGGL(bias_add, dim3(gblk(nOut, B)), dim3(B), 0, stream,
                     out, b_tt, b_bt, b_tb, NTX, NBD);
}